// GridEncoder_47253230191158
// MI455X (gfx1250) — compile-verified
//
#include <hip/hip_runtime.h>
#include <hip/hip_bf16.h>

// instant-NGP hash-grid encode, specialized to:
// INPUT_DIM=3, NUM_LEVELS=16, LEVEL_DIM=2, scale=2.0, BASE_RES=16,
// LOG2_HASHMAP_SIZE=19, ALIGN_CORNERS=false, B=1<<20.
//
// Level l: scale = 16*2^l - 1, side = 16*2^l + 1.
// Levels 0..2 dense (side^3 <= hmap), levels 3..15 hashed (hmap = 2^19).

#define PRIME1 2654435761u
#define PRIME2 805459861u

typedef float v4f __attribute__((ext_vector_type(4)));

__device__ __forceinline__ unsigned lds_byte_off(const void* p) {
    // generic pointers to LDS carry the DS byte offset in the low 32 bits
    return (unsigned)(unsigned long long)p;
}

__global__ __launch_bounds__(256) void grid_encode_47253230191158(
    const float* __restrict__ inp,      // [B,3]
    const float2* __restrict__ emb,     // [n_entries,2] as float2
    float* __restrict__ out,            // [B,32]
    int n)
{
    // offsets[l] into the embedding table (matches _offsets())
    static constexpr unsigned kOffset[16] = {
        0u, 4920u, 40864u, 315496u, 839784u, 1364072u, 1888360u, 2412648u,
        2936936u, 3461224u, 3985512u, 4509800u, 5034088u, 5558376u,
        6082664u, 6606952u};
    static constexpr unsigned kSide[3] = {17u, 33u, 65u};  // dense levels only

    __shared__ float scoord[256 * 3];  // 3 KB coordinate tile

    const int t = threadIdx.x;
    const long long b = (long long)blockIdx.x * 256 + t;
    const bool full_block = ((long long)(blockIdx.x + 1) * 256) <= (long long)n;

    if (full_block) {
        // CDNA5 async bulk copy: 192 lanes x 16B = 3072B = 256 points x 12B.
        if (t < 192) {
            unsigned long long ga = (unsigned long long)inp
                                  + (unsigned long long)blockIdx.x * 3072ull
                                  + (unsigned)(t * 16);
            unsigned la = lds_byte_off(scoord) + (unsigned)(t * 16);
            asm volatile("global_load_async_to_lds_b128 %0, %1, off"
                         :: "v"(la), "v"(ga) : "memory");
        }
        asm volatile("s_wait_asynccnt 0" ::: "memory");
    } else {
        if (b < n) {
            scoord[t * 3 + 0] = inp[b * 3 + 0];
            scoord[t * 3 + 1] = inp[b * 3 + 1];
            scoord[t * 3 + 2] = inp[b * 3 + 2];
        }
    }
    __syncthreads();
    if (b >= n) return;

    // stride-3 LDS reads: bank = (3*t + c) % 64, gcd(3,64)=1 -> conflict-free
    const float cx = scoord[t * 3 + 0];
    const float cy = scoord[t * 3 + 1];
    const float cz = scoord[t * 3 + 2];

    float acc[32];
#pragma unroll
    for (int i = 0; i < 32; ++i) acc[i] = 0.0f;

#pragma unroll
    for (int l = 0; l < 16; ++l) {
        const float scale = (float)(16u << l) - 1.0f;  // exact in f32

        float px = cx * scale + 0.5f;
        float py = cy * scale + 0.5f;
        float pz = cz * scale + 0.5f;
        float gx = floorf(px), gy = floorf(py), gz = floorf(pz);
        float wx1 = px - gx, wy1 = py - gy, wz1 = pz - gz;
        float wx0 = 1.0f - wx1, wy0 = 1.0f - wy1, wz0 = 1.0f - wz1;
        unsigned x0 = (unsigned)gx, y0 = (unsigned)gy, z0 = (unsigned)gz;

        // per-axis index contributions, then fold y/z into 4 combos so the
        // corner loop is 1 op for index + 1 mul for weight + 2 FMA.
        unsigned ix[2], iyz[4];
        if (l < 3) {  // dense: idx = x + y*side + z*side^2 (< hmap, %hmap is identity)
            const unsigned side = kSide[l];
            const unsigned side2 = side * side;
            const unsigned iy0 = y0 * side,  iy1 = iy0 + side;
            const unsigned iz0 = z0 * side2, iz1 = iz0 + side2;
            ix[0] = x0; ix[1] = x0 + 1u;
            iyz[0] = iy0 + iz0; iyz[1] = iy1 + iz0;
            iyz[2] = iy0 + iz1; iyz[3] = iy1 + iz1;
        } else {      // spatial hash with uint32 wraparound; hmap = 2^19
            const unsigned iy0 = y0 * PRIME1, iy1 = iy0 + PRIME1;
            const unsigned iz0 = z0 * PRIME2, iz1 = iz0 + PRIME2;
            ix[0] = x0; ix[1] = x0 + 1u;
            iyz[0] = iy0 ^ iz0; iyz[1] = iy1 ^ iz0;
            iyz[2] = iy0 ^ iz1; iyz[3] = iy1 ^ iz1;
        }
        const float wxs[2] = {wx0, wx1};
        const float wyz[4] = {wy0 * wz0, wy1 * wz0, wy0 * wz1, wy1 * wz1};

        float r0 = 0.0f, r1 = 0.0f;
#pragma unroll
        for (int k = 0; k < 8; ++k) {
            const int bx = k & 1;
            const int byz = k >> 1;  // bz*2 + by
            unsigned idx;
            if (l < 3) idx = ix[bx] + iyz[byz];
            else       idx = (ix[bx] ^ iyz[byz]) & 0x7FFFFu;
            const float w = wxs[bx] * wyz[byz];
            const float2 f = emb[kOffset[l] + idx];  // 8B gather, L2-resident
            r0 += w * f.x;
            r1 += w * f.y;
        }
        acc[2 * l + 0] = r0;
        acc[2 * l + 1] = r1;
    }

    // 8 x b128 NON-TEMPORAL stores: output is write-once streaming (128 MB);
    // NT hint keeps the 57 MB embedding table resident in the 192 MB L2.
    // A wave still covers a contiguous 4 KB region -> full lines in GL2.
    v4f* o4 = (v4f*)(out + b * 32);
#pragma unroll
    for (int i = 0; i < 8; ++i) {
        v4f v = {acc[4 * i + 0], acc[4 * i + 1], acc[4 * i + 2], acc[4 * i + 3]};
        __builtin_nontemporal_store(v, o4 + i);
    }
}

extern "C" void kernel_launch(void* const* d_in, const int* in_sizes, int n_in,
                              void* d_out, int out_size, void* d_ws, size_t ws_size,
                              hipStream_t stream) {
    const float*  inp = (const float*)d_in[0];
    const float2* emb = (const float2*)d_in[1];
    float* out = (float*)d_out;
    const int n = in_sizes[0] / 3;           // B points
    const int blocks = (n + 255) / 256;
    grid_encode_47253230191158<<<blocks, 256, 0, stream>>>(inp, emb, out, n);
    (void)d_ws; (void)ws_size; (void)n_in; (void)out_size;
}